// LuongAttention_6743098654970
// MI455X (gfx1250) — compile-verified
//
#include <hip/hip_runtime.h>
#include <math.h>

typedef __attribute__((ext_vector_type(2))) float v2f;
typedef __attribute__((ext_vector_type(8))) float v8f;

#define BDIM   32
#define SDIM   2048
#define HDIM   1024
#define NCHUNK 8
#define SC     (SDIM / NCHUNK)   // 256 rows of S per chunk
#define TILE   64                // sub-tile rows for online softmax
#define NT     (SC / TILE)       // 4 sub-tiles per chunk

// ---------------------------------------------------------------------------
// Kernel 1: h = hidden @ W_in^T  via V_WMMA_F32_16X16X4_F32 (exact f32 path).
// One wave per 16x16 output tile. A = hidden[16x4 slice], B = W_in rows
// (since h[m,n] = sum_k hidden[m,k] * W_in[n,k], both operands read k-contig).
// f32 16x4 A layout: lanes 0-15 hold M=0..15 {K=0,K=1}, lanes 16-31 {K=2,K=3}.
// B (4x16) mirrors: lanes 0-15 = K=0..1 per VGPR, lanes 16-31 = K=2..3, N=lane%16.
// ---------------------------------------------------------------------------
__global__ __launch_bounds__(32) void k_gemm_h(const float* __restrict__ hidden,
                                               const float* __restrict__ W_in,
                                               float* __restrict__ h_out) {
    const int nbase = blockIdx.x * 16;
    const int mbase = blockIdx.y * 16;
    const int lane  = threadIdx.x;         // 0..31 (wave32)
    const int r16   = lane & 15;
    const int kpair = (lane >> 4) * 2;     // 0 or 2

    const float* Arow = hidden + (size_t)(mbase + r16) * HDIM + kpair;
    const float* Brow = W_in   + (size_t)(nbase + r16) * HDIM + kpair;

    v8f c = {};
    for (int k0 = 0; k0 < HDIM; k0 += 4) {
        v2f a = *(const v2f*)(Arow + k0);
        v2f b = *(const v2f*)(Brow + k0);
        c = __builtin_amdgcn_wmma_f32_16x16x4_f32(
                /*neg_a=*/false, a, /*neg_b=*/false, b,
                /*c_mod=*/(short)0, c, /*reuse_a=*/false, /*reuse_b=*/false);
    }

    // C/D layout: VGPR r -> M = r (lanes 0-15) / r+8 (lanes 16-31), N = lane%16
    const int mhi = (lane >> 4) * 8;
    const int n   = nbase + r16;
#pragma unroll
    for (int r = 0; r < 8; ++r) {
        h_out[(size_t)(mbase + mhi + r) * HDIM + n] = c[r];
    }
}

// ---------------------------------------------------------------------------
// Kernel 2: fused scores + online softmax + context partials.
// One block per (batch, S-chunk). Single HBM pass over encoder_outputs:
// each 64-row sub-tile is streamed for scores, then immediately re-read
// (L2-resident) for the weighted context accumulation with flash-style
// rescaling. Raw scores are written to the attn region of d_out; kernel 3
// normalizes them in place.
// ---------------------------------------------------------------------------
__global__ __launch_bounds__(256) void k_attn_partial(
    const float* __restrict__ enc,    // [32,2048,1024]
    const float* __restrict__ h_buf,  // [32,1024]
    float* __restrict__ raw_scores,   // [32,2048] (attn region of d_out)
    float* __restrict__ pm,           // [32*8] chunk max
    float* __restrict__ pl,           // [32*8] chunk sum
    float* __restrict__ ppc)          // [32*8,1024] chunk partial context
{
    __shared__ float hb[HDIM];
    __shared__ float sc_tile[TILE];
    __shared__ float w_tile[TILE];

    const int b    = blockIdx.y;
    const int cch  = blockIdx.x;
    const int t    = threadIdx.x;
    const int lane = t & 31;
    const int wave = t >> 5;

    for (int i = t; i < HDIM; i += 256) hb[i] = h_buf[(size_t)b * HDIM + i];
    __syncthreads();

    const float*  encB = enc + (size_t)b * SDIM * HDIM;
    const float4* hb4  = (const float4*)hb;
    const int     s0   = cch * SC;
    const int     hcol = t * 4;        // 4 contiguous H columns owned in phase 2

    float  M_run = -3.4e38f, L_run = 0.f;
    float4 pc = make_float4(0.f, 0.f, 0.f, 0.f);

    for (int tile = 0; tile < NT; ++tile) {
        const int sb = s0 + tile * TILE;

        // -------- phase 1: scores for 64 rows (wave w -> rows w, w+8, ...)
        for (int i = 0; i < TILE / 8; ++i) {
            const int r = wave + 8 * i;
            const int s = sb + r;
            const float4* erow = (const float4*)(encB + (size_t)s * HDIM);
            float acc = 0.f;
#pragma unroll
            for (int j = 0; j < 8; ++j) {           // 32 lanes x 8 float4 = 1024
                float4 e = erow[lane + 32 * j];
                float4 h = hb4[lane + 32 * j];
                acc += e.x * h.x + e.y * h.y + e.z * h.z + e.w * h.w;
            }
            for (int off = 16; off > 0; off >>= 1)  // wave32 reduction
                acc += __shfl_xor(acc, off, 32);
            if (lane == 0) {
                sc_tile[r] = acc;
                raw_scores[(size_t)b * SDIM + s] = acc;  // raw; normalized later
            }
        }
        __syncthreads();

        // -------- online softmax update (uniform across threads)
        float mt = -3.4e38f;
#pragma unroll 8
        for (int ss = 0; ss < TILE; ++ss) mt = fmaxf(mt, sc_tile[ss]);
        const float newM  = fmaxf(M_run, mt);
        const float scale = __expf(M_run - newM);

        if (t < TILE) w_tile[t] = __expf(sc_tile[t] - newM);
        __syncthreads();

        float sumw = 0.f;
#pragma unroll 8
        for (int ss = 0; ss < TILE; ++ss) sumw += w_tile[ss];

        // -------- phase 2: context accumulation (tile re-read hits L2)
        pc.x *= scale; pc.y *= scale; pc.z *= scale; pc.w *= scale;
        const float4* ecol = (const float4*)(encB + (size_t)sb * HDIM + hcol);
        for (int ss = 0; ss < TILE; ++ss) {
            float4 e = ecol[ss * (HDIM / 4)];
            const float w = w_tile[ss];
            pc.x += w * e.x; pc.y += w * e.y; pc.z += w * e.z; pc.w += w * e.w;
            if (tile + 1 < NT)   // warm L2 for next sub-tile (global_prefetch_b8)
                __builtin_prefetch((const void*)(ecol + (ss + TILE) * (HDIM / 4)), 0, 1);
        }
        L_run = L_run * scale + sumw;
        M_run = newM;
        __syncthreads();   // sc_tile/w_tile reused next sub-tile
    }

    const int pidx = b * NCHUNK + cch;
    if (t == 0) { pm[pidx] = M_run; pl[pidx] = L_run; }
    *(float4*)(ppc + (size_t)pidx * HDIM + hcol) = pc;
}

// ---------------------------------------------------------------------------
// Kernel 3: merge the 8 chunk partials per batch; write context and
// normalize raw scores into softmax weights in place.
// ---------------------------------------------------------------------------
__global__ __launch_bounds__(256) void k_combine(
    const float* __restrict__ pm, const float* __restrict__ pl,
    const float* __restrict__ ppc,
    float* __restrict__ context,   // [32,1024]
    float* __restrict__ attn)      // [32,2048] raw in -> normalized out
{
    const int b = blockIdx.x;
    const int t = threadIdx.x;

    float M = -3.4e38f;
#pragma unroll
    for (int c = 0; c < NCHUNK; ++c) M = fmaxf(M, pm[b * NCHUNK + c]);
    float L = 0.f;
    float f[NCHUNK];
#pragma unroll
    for (int c = 0; c < NCHUNK; ++c) {
        f[c] = __expf(pm[b * NCHUNK + c] - M);
        L += pl[b * NCHUNK + c] * f[c];
    }
    const float invL = 1.0f / L;

    const int hcol = t * 4;
    float4 acc = make_float4(0.f, 0.f, 0.f, 0.f);
#pragma unroll
    for (int c = 0; c < NCHUNK; ++c) {
        const float4 p = *(const float4*)(ppc + (size_t)(b * NCHUNK + c) * HDIM + hcol);
        acc.x += f[c] * p.x; acc.y += f[c] * p.y;
        acc.z += f[c] * p.z; acc.w += f[c] * p.w;
    }
    acc.x *= invL; acc.y *= invL; acc.z *= invL; acc.w *= invL;
    *(float4*)(context + (size_t)b * HDIM + hcol) = acc;

#pragma unroll
    for (int i = 0; i < SDIM / 256; ++i) {
        const size_t idx = (size_t)b * SDIM + t + 256 * i;
        const float raw = attn[idx];
        attn[idx] = __expf(raw - M) * invL;
    }
}

// ---------------------------------------------------------------------------
extern "C" void kernel_launch(void* const* d_in, const int* in_sizes, int n_in,
                              void* d_out, int out_size, void* d_ws, size_t ws_size,
                              hipStream_t stream) {
    const float* hidden = (const float*)d_in[0];   // [32,1024]
    const float* enc    = (const float*)d_in[1];   // [32,2048,1024]
    const float* W_in   = (const float*)d_in[2];   // [1024,1024]

    float* out     = (float*)d_out;
    float* context = out;                       // 32*1024
    float* attn    = out + BDIM * HDIM;         // 32*2048

    float* ws    = (float*)d_ws;
    float* h_buf = ws;                          // 32768 floats
    float* pm    = h_buf + BDIM * HDIM;         // 256
    float* pl    = pm + BDIM * NCHUNK;          // 256
    float* ppc   = pl + BDIM * NCHUNK;          // 32*8*1024 floats

    k_gemm_h<<<dim3(HDIM / 16, BDIM / 16), 32, 0, stream>>>(hidden, W_in, h_buf);
    k_attn_partial<<<dim3(NCHUNK, BDIM), 256, 0, stream>>>(enc, h_buf, attn, pm, pl, ppc);
    k_combine<<<BDIM, 256, 0, stream>>>(pm, pl, ppc, context, attn);
}